// CrossNetwork_47021301957253
// MI455X (gfx1250) — compile-verified
//
#include <hip/hip_runtime.h>

// CrossNetwork collapsed to: C = X @ W^T (WMMA f32 16x16x4), per-row scalar
// recurrence for alpha3, then out = x*alpha3 + beta3. Single pass over X via
// a 16x2048 f32 tile staged in LDS with CDNA5 async global->LDS loads.

#define DIMD   2048
#define ROWS   16
#define PITCH  2052          // 2048 + 4 pad -> bank shift of 4 per row (conflict-free)
#define NTHR   256           // 8 wave32 waves
#define KCHUNK (DIMD / 8)    // 256 columns per wave

typedef __attribute__((ext_vector_type(2))) float v2f;
typedef __attribute__((ext_vector_type(8))) float v8f;

// ---- CDNA5 async global->LDS copy (ASYNCcnt-tracked, no VGPR data) --------
__device__ __forceinline__ void async_g2l_b128(uint32_t lds_byte_addr,
                                               const void* gaddr) {
  asm volatile("global_load_async_to_lds_b128 %0, %1, off"
               :
               : "v"(lds_byte_addr), "v"(gaddr)
               : "memory");
}
__device__ __forceinline__ void wait_asynccnt0() {
  asm volatile("s_wait_asynccnt 0" ::: "memory");
}

// ---------------------------------------------------------------------------
// Setup: ws[0]=gamma1=<b0,w1>, ws[1]=gamma2=<b0+b1,w2>, ws[4..4+D)=beta3=b0+b1+b2
// ---------------------------------------------------------------------------
__global__ __launch_bounds__(NTHR) void cross_setup_kernel(
    const float* __restrict__ W, const float* __restrict__ b,
    float* __restrict__ ws) {
  __shared__ float red1[NTHR];
  __shared__ float red2[NTHR];
  const int tid = threadIdx.x;
  float g1 = 0.f, g2 = 0.f;
  for (int k = tid; k < DIMD; k += NTHR) {
    const float b0 = b[0 * DIMD + k];
    const float b1 = b[1 * DIMD + k];
    const float b2 = b[2 * DIMD + k];
    ws[4 + k] = b0 + b1 + b2;
    g1 += b0 * W[1 * DIMD + k];
    g2 += (b0 + b1) * W[2 * DIMD + k];
  }
  red1[tid] = g1;
  red2[tid] = g2;
  __syncthreads();
  for (int s = NTHR / 2; s > 0; s >>= 1) {
    if (tid < s) { red1[tid] += red1[tid + s]; red2[tid] += red2[tid + s]; }
    __syncthreads();
  }
  if (tid == 0) { ws[0] = red1[0]; ws[1] = red2[0]; }
}

// ---------------------------------------------------------------------------
// Main: one block = 16 rows of X. Async-stage tile in LDS, WMMA the 3 dot
// products, scalar recurrence, axpy out of LDS. One HBM read, one HBM write.
// ---------------------------------------------------------------------------
__global__ __launch_bounds__(NTHR) void cross_main_kernel(
    const float* __restrict__ x, const float* __restrict__ W,
    const float* __restrict__ ws, float* __restrict__ out) {
  __shared__ __align__(16) float lx[ROWS * PITCH];   // 131,328 B
  __shared__ __align__(16) float lw[4 * PITCH];      //  32,832 B (row 3 = zeros)
  __shared__ float red[3 * 16 * 8];                  //   1,536 B
  __shared__ float alphaS[ROWS];

  const int tid  = threadIdx.x;
  const int wave = tid >> 5;     // 0..7
  const int lane = tid & 31;
  const int half = lane >> 4;    // 0/1  (K sub-pair)
  const int lm   = lane & 15;    // A: row M; B: col N

  // ---- Stage X tile via async global->LDS (16 rows, contiguous source) ----
  // AS3->generic on AMDGPU is {SHARED_BASE, offset}: low 32 bits of the
  // generic pointer ARE the LDS byte address the async VDST operand wants.
  const uint32_t lxbase = (uint32_t)(uintptr_t)(const void*)&lx[0];
  const size_t tile_base = (size_t)blockIdx.x * (size_t)(ROWS * DIMD);
  const float4* xsrc = (const float4*)(x + tile_base);
  for (int i = tid; i < ROWS * DIMD / 4; i += NTHR) {
    const int row = i / (DIMD / 4);
    const int col = (i % (DIMD / 4)) * 4;
    async_g2l_b128(lxbase + (uint32_t)(row * PITCH + col) * 4u, xsrc + i);
  }
  // ---- Stage W (3 rows) + a zero row so every lane has a valid B address --
  for (int i = tid; i < 3 * DIMD / 4; i += NTHR) {
    const float4 v = ((const float4*)W)[i];
    const int row = i / (DIMD / 4);
    const int col = (i % (DIMD / 4)) * 4;
    *(float4*)&lw[row * PITCH + col] = v;
  }
  for (int k = tid * 4; k < DIMD; k += NTHR * 4)
    *(float4*)&lw[3 * PITCH + k] = float4{0.f, 0.f, 0.f, 0.f};
  wait_asynccnt0();              // own async transfers landed in LDS
  __syncthreads();               // everyone's transfers visible

  // ---- WMMA accumulation: C += A(16x4 of X) * B(4x16 of W^T) --------------
  // A lane layout (ISA 7.12.2): lane holds X[lm][k + 2*half + {0,1}]
  // B lane layout (mirrored):   lane holds W[lm][k + 2*half + {0,1}]  (lm<3 real)
  const int wrow = (lm < 3) ? lm : 3;                // zero row for pad lanes
  const float* ax = &lx[lm * PITCH + wave * KCHUNK + 2 * half];
  const float* bw = &lw[wrow * PITCH + wave * KCHUNK + 2 * half];

  v8f acc = {0.f, 0.f, 0.f, 0.f, 0.f, 0.f, 0.f, 0.f};
#pragma unroll 8
  for (int k = 0; k < KCHUNK; k += 4) {
    v2f a, bb;
    a.x  = ax[k];
    a.y  = ax[k + 1];
    bb.x = bw[k];
    bb.y = bw[k + 1];
    acc = __builtin_amdgcn_wmma_f32_16x16x4_f32(
        /*neg_a=*/false, a, /*neg_b=*/false, bb,
        /*c_mod=*/(short)0, acc, /*reuse_a=*/false, /*reuse_b=*/false);
  }

  // ---- Cross-wave reduction of C columns n=0..2 ---------------------------
  // C layout: lane = n + 16*(m/8), vgpr v = m%8  ->  here m = v + 8*half, n = lm
  if (lm < 3) {
#pragma unroll
    for (int v = 0; v < 8; ++v) {
      const int m = v + 8 * half;
      red[(lm * 16 + m) * 8 + wave] = acc[v];
    }
  }
  __syncthreads();

  // ---- Per-row scalar recurrence ------------------------------------------
  if (tid < ROWS) {
    float c[3];
#pragma unroll
    for (int n = 0; n < 3; ++n) {
      float s = 0.f;
#pragma unroll
      for (int w = 0; w < 8; ++w) s += red[(n * 16 + tid) * 8 + w];
      c[n] = s;
    }
    const float g1 = ws[0];
    const float g2 = ws[1];
    const float a1 = 1.f + c[0];
    const float a2 = a1 * (1.f + c[1]) + g1;
    const float a3 = a2 * (1.f + c[2]) + g2;
    alphaS[tid] = a3;
  }
  __syncthreads();

  // ---- out = x * alpha3[row] + beta3[col]  (x from LDS, coalesced store) --
  const float* beta = ws + 4;                        // 16B aligned
  float4* dst = (float4*)(out + tile_base);
  for (int i = tid; i < ROWS * DIMD / 4; i += NTHR) {
    const int row = i / (DIMD / 4);
    const int col = (i % (DIMD / 4)) * 4;
    const float4 xv = *(const float4*)&lx[row * PITCH + col];
    const float4 bv = *(const float4*)&beta[col];
    const float  al = alphaS[row];
    float4 o;
    o.x = xv.x * al + bv.x;
    o.y = xv.y * al + bv.y;
    o.z = xv.z * al + bv.z;
    o.w = xv.w * al + bv.w;
    dst[i] = o;
  }
}

// ---------------------------------------------------------------------------
extern "C" void kernel_launch(void* const* d_in, const int* in_sizes, int n_in,
                              void* d_out, int out_size, void* d_ws, size_t ws_size,
                              hipStream_t stream) {
  const float* x = (const float*)d_in[0];   // [16384, 2048]
  const float* W = (const float*)d_in[1];   // [3, 2048]
  const float* b = (const float*)d_in[2];   // [3, 2048]
  float* out = (float*)d_out;               // [16384, 2048]
  float* ws  = (float*)d_ws;                // [0]=g1 [1]=g2 [4..4+2048)=beta3

  cross_setup_kernel<<<1, NTHR, 0, stream>>>(W, b, ws);

  const int nblocks = 16384 / ROWS;         // 1024
  cross_main_kernel<<<nblocks, NTHR, 0, stream>>>(x, W, ws, out);
}